// _OuterModel_57655640981802
// MI455X (gfx1250) — compile-verified
//
#include <hip/hip_runtime.h>

// out[n,e] = sum_d x[n,d] * W[mm[n], e, d]   (N=1M tokens, H=64, 3 modalities)
//
// Memory-bound kernel (~520MB moved, ~22us floor @ 23.3 TB/s). Strategy:
//  - transpose the GEMM: W rows are the WMMA A-matrix, kept RESIDENT IN VGPRs
//    (48KB of weights total -> 64 VGPRs per wave for its slice),
//  - 6 waves per workgroup = (3 modalities) x (2 thirty-two-wide e-halves),
//  - every wave computes its modality's candidate for a 16-token tile with
//    V_WMMA_F32_16X16X4_F32 (full fp32 precision) and stores only the output
//    columns whose token modality matches (EXEC-predicated, disjoint columns),
//  - x tile read by the 3 modality waves of one WGP -> HBM reads it once,
//    WGP-scope prefetch of the next tile, non-temporal streaming stores.

typedef __attribute__((ext_vector_type(2))) float v2f;
typedef __attribute__((ext_vector_type(4))) float v4f;
typedef __attribute__((ext_vector_type(8))) float v8f;

#define HID 64
#define NMOD 3
#define TILES_PER_WG 16          // 16-token tiles handled per workgroup
#define WAVES_PER_WG 6           // (modality 0..2) x (e-half 0..1)

__global__ __launch_bounds__(WAVES_PER_WG * 32) void
moe_linear_f32_wmma(const float* __restrict__ x,
                    const float* __restrict__ W,
                    const int*   __restrict__ mm,
                    float* __restrict__ out,
                    int ntok)
{
    const int lane = threadIdx.x & 31;
    const int wave = threadIdx.x >> 5;
    const int mod  = wave >> 1;          // 0..2 : modality this wave evaluates
    const int eh   = wave & 1;           // which 32-wide half of the 64 outputs
    const int half = lane >> 4;          // 0: lanes 0-15, 1: lanes 16-31
    const int l16  = lane & 15;

    // ---- Persistent A fragments: A[M=e][K=d], 2 e-tiles of 16 rows.
    // ISA A-layout (16x4 fp32): lane holds row M=lane%16;
    // VGPR j = A[M][kc*4 + 2*half + j]  -> one aligned 8B load per fragment.
    v2f a[2][16];
#pragma unroll
    for (int et = 0; et < 2; ++et) {
        const float* wr = W + ((size_t)mod * HID + (size_t)(eh * 32 + et * 16 + l16)) * HID
                            + 2 * half;
#pragma unroll
        for (int kc = 0; kc < 16; ++kc)
            a[et][kc] = *(const v2f*)(wr + kc * 4);
    }

    for (int tile = 0; tile < TILES_PER_WG; ++tile) {
        const int t0 = ((int)blockIdx.x * TILES_PER_WG + tile) * 16;
        if (t0 >= ntok) return;
        const int t = t0 + l16;                       // this lane's token column
        const float* xr = x + (size_t)t * HID + 2 * half;

        // ---- B fragments: B[K=d][N=t] = x[t][d].
        // lane holds column N=lane%16; VGPR j = B[kc*4 + 2*half + j][N].
        v2f b[16];
#pragma unroll
        for (int kc = 0; kc < 16; ++kc)
            b[kc] = *(const v2f*)(xr + kc * 4);

        if (tile + 1 < TILES_PER_WG)
            __builtin_prefetch(xr + 16 * HID, 0, 3);  // WGP-scope: fill all levels

        // ---- outT[16e x 16t] += A(16x4) x B(4x16), K=64 in 16 chunks of 4.
        v8f acc[2] = {};
#pragma unroll
        for (int kc = 0; kc < 16; ++kc) {
            acc[0] = __builtin_amdgcn_wmma_f32_16x16x4_f32(
                false, a[0][kc], false, b[kc], (short)0, acc[0], false, false);
            acc[1] = __builtin_amdgcn_wmma_f32_16x16x4_f32(
                false, a[1][kc], false, b[kc], (short)0, acc[1], false, false);
        }

        // ---- Predicated store: only the wave whose modality matches token t
        // writes column t. D-layout: lane = column t, VGPR v = row e0+v+8*half
        // -> rows are consecutive floats: two 16B stores per e-tile.
        // Streaming output -> non-temporal (don't pollute 192MB L2).
        if (mm[t] == mod) {
            float* orow = out + (size_t)t * HID + eh * 32 + 8 * half;
#pragma unroll
            for (int et = 0; et < 2; ++et) {
                v4f lo = {acc[et][0], acc[et][1], acc[et][2], acc[et][3]};
                v4f hi = {acc[et][4], acc[et][5], acc[et][6], acc[et][7]};
                __builtin_nontemporal_store(lo, (v4f*)(orow + et * 16));
                __builtin_nontemporal_store(hi, (v4f*)(orow + et * 16 + 4));
            }
        }
    }
}

extern "C" void kernel_launch(void* const* d_in, const int* in_sizes, int n_in,
                              void* d_out, int out_size, void* d_ws, size_t ws_size,
                              hipStream_t stream)
{
    const float* x  = (const float*)d_in[0];   // [N, 64] fp32
    const float* W  = (const float*)d_in[1];   // [3, 64, 64] fp32
    const int*   mm = (const int*)d_in[2];     // [N] modality ids
    float*       out = (float*)d_out;          // [N, 64] fp32

    const int ntok  = in_sizes[0] / HID;
    const int tiles = (ntok + 15) / 16;
    const int wgs   = (tiles + TILES_PER_WG - 1) / TILES_PER_WG;

    hipLaunchKernelGGL(moe_linear_f32_wmma, dim3(wgs), dim3(WAVES_PER_WG * 32),
                       0, stream, x, W, mm, out, ntok);
}